// unit_ctrgcn_sparse_32006096290316
// MI455X (gfx1250) — compile-verified
//
#include <hip/hip_runtime.h>

// Problem constants (from reference)
#define Nn 16
#define Cc 64
#define Tt 512
#define Vv 25
#define Rr 8
#define Oo 64
#define Ss 3

// Padded attention-matrix dims (zero padding => no masking inside WMMA loops)
#define UP 28   // u padded (25 -> 28, multiple of 4 for K-chunks)
#define VP 32   // v padded (25 -> 32, two 16-wide N tiles)

// Main-kernel tiling
#define TTILE 8
#define PP 28                 // pixels per t row (u padded)
#define PIX (TTILE * PP)      // 224 pixels per tile
#define PADC 66               // LDS row stride for xt[p][c]
#define PADP 226              // LDS row stride for x3t[o][p]

typedef __attribute__((ext_vector_type(2))) float v2f;
typedef __attribute__((ext_vector_type(8))) float v8f;

__device__ __forceinline__ v8f wmma4(v2f a, v2f b, v8f c) {
  // D = A(16x4) * B(4x16) + C, all f32.  8-arg form:
  // (neg_a, A, neg_b, B, c_mod, C, reuse_a, reuse_b)
  return __builtin_amdgcn_wmma_f32_16x16x4_f32(false, a, false, b, (short)0, c,
                                               false, false);
}

__device__ __forceinline__ float wmask(float w, float thr) {
  return (fabsf(w) > thr) ? w : 0.0f;
}

// ---------------------------------------------------------------------------
// K1: xbar[n][c][v] = mean over t of x[n][c][t][v]
// grid = Nn*Cc blocks, 256 threads
// ---------------------------------------------------------------------------
__global__ void k_xbar(const float* __restrict__ x, float* __restrict__ xbar) {
  int nc = blockIdx.x;
  int v = threadIdx.x & 31;
  int grp = threadIdx.x >> 5;  // 8 groups of 32
  __shared__ float sm[8 * 32];
  float acc = 0.0f;
  if (v < Vv) {
    const float* p = x + (size_t)nc * Tt * Vv + v;
    for (int t = grp; t < Tt; t += 8) acc += p[(size_t)t * Vv];
  }
  sm[grp * 32 + v] = acc;
  __syncthreads();
  if (grp == 0 && v < Vv) {
    float tot = 0.0f;
#pragma unroll
    for (int g = 0; g < 8; ++g) tot += sm[g * 32 + v];
    xbar[nc * Vv + v] = tot * (1.0f / Tt);
  }
}

// ---------------------------------------------------------------------------
// K2: a_pad[s][n][o][u:UP][v:VP] =
//     (sum_r mask(w4[s][o][r]) * tanh(x1[s][n][r][u] - x2[s][n][r][v])
//      + b4[s][o]) * alpha + A[s][u][v],   zero in padding
// with x1 = mask(w1) @ xbar + b1, x2 = mask(w2) @ xbar + b2
// grid = Ss*Nn blocks, 256 threads
// ---------------------------------------------------------------------------
__global__ void k_amat(const float* __restrict__ xbar,
                       const float* __restrict__ A,
                       const float* __restrict__ alpha,
                       const float* __restrict__ w1, const float* __restrict__ b1,
                       const float* __restrict__ w2, const float* __restrict__ b2,
                       const float* __restrict__ w4, const float* __restrict__ b4,
                       const int* __restrict__ sp,
                       float* __restrict__ a_pad) {
  int s = blockIdx.x >> 4;
  int n = blockIdx.x & 15;
  float thr = (float)sp[0];
  __shared__ float sx1[Rr * Vv];
  __shared__ float sx2[Rr * Vv];
  __shared__ float stt[Rr * Vv * Vv];
  int tid = threadIdx.x;
  if (tid < Rr * Vv) {
    int r = tid / Vv, v = tid % Vv;
    float a1 = b1[s * Rr + r];
    float a2 = b2[s * Rr + r];
    for (int c = 0; c < Cc; ++c) {
      float xb = xbar[(n * Cc + c) * Vv + v];
      a1 += wmask(w1[(s * Rr + r) * Cc + c], thr) * xb;
      a2 += wmask(w2[(s * Rr + r) * Cc + c], thr) * xb;
    }
    sx1[tid] = a1;
    sx2[tid] = a2;
  }
  __syncthreads();
  for (int idx = tid; idx < Rr * Vv * Vv; idx += 256) {
    int r = idx / (Vv * Vv), rem = idx % (Vv * Vv);
    int u = rem / Vv, v = rem % Vv;
    stt[idx] = tanhf(sx1[r * Vv + u] - sx2[r * Vv + v]);
  }
  __syncthreads();
  float al = alpha[0];
  for (int idx = tid; idx < Oo * UP * VP; idx += 256) {
    int o = idx / (UP * VP), rem = idx % (UP * VP);
    int u = rem / VP, v = rem % VP;
    float val = 0.0f;
    if (u < Vv && v < Vv) {
      float acc = 0.0f;
#pragma unroll
      for (int r = 0; r < Rr; ++r)
        acc += wmask(w4[(s * Oo + o) * Rr + r], thr) * stt[r * Vv * Vv + u * Vv + v];
      val = (acc + b4[s * Oo + o]) * al + A[s * Vv * Vv + u * Vv + v];
    }
    a_pad[((((size_t)s * Nn + n) * Oo + o) * UP + u) * VP + v] = val;
  }
}

// ---------------------------------------------------------------------------
// K3 (fused, WMMA): per (n, t-tile of 8):
//   load x tile once into LDS; for each subset s:
//     Stage A: x3t[o][p] = sum_c mask(w3)[o][c]*xt[p][c] + b3[o]   (WMMA f32)
//     Stage B: accz[c][t][v] += sum_u x3t[c][t*28+u] * a_pad[s,n,c,u,v] (WMMA)
//   Stage C: out = relu(accz + x residual)
// grid = Nn * (Tt/TTILE) = 1024 blocks, 512 threads (16 waves)
// ---------------------------------------------------------------------------
__global__ void __launch_bounds__(512)
k_main(const float* __restrict__ x,
       const float* __restrict__ w3, const float* __restrict__ b3,
       const float* __restrict__ a_pad, const int* __restrict__ sp,
       float* __restrict__ out) {
  __shared__ float xt[PIX * PADC];   // [p][c], p = tl*28 + u (u>=25 zeroed)
  __shared__ float x3t[Oo * PADP];   // [o][p]

  int n = blockIdx.x >> 6;    // batch
  int tb = blockIdx.x & 63;   // t tile
  int t0 = tb * TTILE;
  int tid = threadIdx.x;
  int wave = tid >> 5;
  int lane = tid & 31;
  int lo = lane & 15;
  int hi = lane >> 4;
  float thr = (float)sp[0];

  // cooperative load of the x tile (also serves as the residual)
  for (int idx = tid; idx < PIX * Cc; idx += 512) {
    int c = idx / PIX;
    int p = idx % PIX;
    int u = p % PP, tl = p / PP;
    float val = 0.0f;
    if (u < Vv) val = x[(((size_t)n * Cc + c) * Tt + (t0 + tl)) * Vv + u];
    xt[p * PADC + c] = val;
  }
  __syncthreads();

  v8f accz[8];
#pragma unroll
  for (int j = 0; j < 8; ++j) {
    v8f z = {0.f, 0.f, 0.f, 0.f, 0.f, 0.f, 0.f, 0.f};
    accz[j] = z;
  }

  for (int s = 0; s < Ss; ++s) {
    // ---- Stage A: 64x64 GEMM over 224 pixels, tiles (ot, pt): 4*14 = 56 jobs
#pragma unroll
    for (int j = 0; j < 4; ++j) {
      int job = wave + 16 * j;
      if (job < 56) {  // wave-uniform branch
        int ot = job / 14, pt = job % 14;
        int p = pt * 16 + lo;
        v8f acc = {0.f, 0.f, 0.f, 0.f, 0.f, 0.f, 0.f, 0.f};
#pragma unroll
        for (int kk = 0; kk < 16; ++kk) {
          int c0 = kk * 4 + 2 * hi;
          const float* wp = w3 + (((size_t)s * Oo + ot * 16 + lo) * Cc + c0);
          v2f av;
          av.x = wmask(wp[0], thr);
          av.y = wmask(wp[1], thr);
          v2f bv = *(const v2f*)(&xt[p * PADC + c0]);
          acc = wmma4(av, bv, acc);
        }
#pragma unroll
        for (int g = 0; g < 8; ++g) {
          int o = ot * 16 + g + 8 * hi;
          x3t[o * PADP + p] = acc[g] + b3[s * Oo + o];
        }
      }
    }
    __syncthreads();

    // ---- Stage B: per-channel [8t x 25u] @ [25u x 25v], jobs (c, vtile): 128
#pragma unroll
    for (int j = 0; j < 8; ++j) {
      int job = wave + 16 * j;
      int c = job >> 1, vt = job & 1;
      const float* ap = a_pad + (((size_t)s * Nn + n) * Oo + c) * (UP * VP);
      v8f az = accz[j];
#pragma unroll
      for (int kk = 0; kk < 7; ++kk) {
        int u0 = kk * 4 + 2 * hi;
        // A rows are t (only 0..7 valid): clamp lo into [0,8) -> rows 8..15 dup
        v2f av = *(const v2f*)(&x3t[c * PADP + (lo & 7) * PP + u0]);
        v2f bv;
        bv.x = ap[u0 * VP + vt * 16 + lo];
        bv.y = ap[(u0 + 1) * VP + vt * 16 + lo];
        az = wmma4(av, bv, az);
      }
      accz[j] = az;
    }
    __syncthreads();  // before next subset's Stage A rewrites x3t
  }

  // ---- Stage C: residual + ReLU + store (valid rows live in lanes hi==0)
#pragma unroll
  for (int j = 0; j < 8; ++j) {
    int job = wave + 16 * j;
    int c = job >> 1, vt = job & 1;
    int v = vt * 16 + lo;
    if (hi == 0 && v < Vv) {
#pragma unroll
      for (int g = 0; g < 8; ++g) {
        int t = t0 + g;
        float res = xt[(g * PP + v) * PADC + c];
        float z = accz[j][g] + res;
        out[(((size_t)n * Cc + c) * Tt + t) * Vv + v] = fmaxf(z, 0.0f);
      }
    }
  }
}

// ---------------------------------------------------------------------------
extern "C" void kernel_launch(void* const* d_in, const int* in_sizes, int n_in,
                              void* d_out, int out_size, void* d_ws, size_t ws_size,
                              hipStream_t stream) {
  (void)in_sizes; (void)n_in; (void)out_size; (void)ws_size;
  const float* x     = (const float*)d_in[0];
  const float* A     = (const float*)d_in[1];
  const float* alpha = (const float*)d_in[2];
  const float* w1    = (const float*)d_in[3];
  const float* b1    = (const float*)d_in[4];
  const float* w2    = (const float*)d_in[5];
  const float* b2    = (const float*)d_in[6];
  const float* w3    = (const float*)d_in[7];
  const float* b3    = (const float*)d_in[8];
  const float* w4    = (const float*)d_in[9];
  const float* b4    = (const float*)d_in[10];
  const int*   sp    = (const int*)d_in[11];
  float* out = (float*)d_out;

  float* xbar  = (float*)d_ws;                       // Nn*Cc*Vv floats
  float* a_pad = xbar + (size_t)Nn * Cc * Vv;        // Ss*Nn*Oo*UP*VP floats

  k_xbar<<<Nn * Cc, 256, 0, stream>>>(x, xbar);
  k_amat<<<Ss * Nn, 256, 0, stream>>>(xbar, A, alpha, w1, b1, w2, b2, w4, b4,
                                      sp, a_pad);
  k_main<<<Nn * (Tt / TTILE), 512, 0, stream>>>(x, w3, b3, a_pad, sp, out);
}